// GraphAttentionLayer_14439680049610
// MI455X (gfx1250) — compile-verified
//
#include <hip/hip_runtime.h>
#include <hip/hip_bf16.h>

// ---------------------------------------------------------------------------
// GAT layer for MI455X (gfx1250, wave32, WMMA)
//   h = x @ W                      -> fp32 WMMA 16x16x4
//   f1/f2 = h @ a1/a2 + b          -> per-row dot
//   e = leaky(adj*f1[src] + adj*f2[dst])
//   softmax over segments of src, out[src] += attn * h[dst], + bias
// ---------------------------------------------------------------------------

#define N_NODES 100000
#define N_EDGES 3200000
#define F_IN    256
#define F_OUT   64
#define ALPHA   0.2f

typedef float v2f __attribute__((ext_vector_type(2)));
typedef float v8f __attribute__((ext_vector_type(8)));

// -------------------------------------------------------------------- GEMM --
// 1024 threads = 32 waves. Block stages W in LDS in K-pair-interleaved layout:
//   Wl[pair = k/2][col][k&1], pair-row stride 160 dwords (128 data + 32 pad).
//  - B frag (W[k..k+1][col]) = ONE contiguous ds_load_b64 into an aligned
//    VGPR pair: no repacking movs in the hot loop.
//  - Stride 160 % 64 banks == 32: lanes 0-15 (K-half 0) hit banks 0..31,
//    lanes 16-31 (K-half 2, pair+1) hit banks 32..63 -> conflict-free.
// Each wave owns 16 rows and computes all 4 column tiles (16x64 slab).
#define PAIR_STRIDE 160                      // dwords per K-pair row in LDS

__global__ __launch_bounds__(1024)
void gat_gemm_wmma(const float* __restrict__ x, const float* __restrict__ W,
                   float* __restrict__ h, int nrows) {
    __shared__ float Wl[(F_IN / 2) * PAIR_STRIDE];   // 128*160*4B = 80 KB
    const int tid = threadIdx.x;

    // cooperative staging with K-pair interleave swizzle
    {
        const float4* Wg = (const float4*)W;
        for (int i = tid; i < (F_IN * F_OUT) / 4; i += 1024) {
            const float4 w = Wg[i];
            const int k = i >> 4;            // W row (16 float4 per 64-col row)
            const int c = (i & 15) * 4;      // starting column
            float* dst = Wl + (k >> 1) * PAIR_STRIDE + (k & 1);
            dst[(c + 0) * 2] = w.x;
            dst[(c + 1) * 2] = w.y;
            dst[(c + 2) * 2] = w.z;
            dst[(c + 3) * 2] = w.w;
        }
    }
    __syncthreads();

    const int wave = tid >> 5;
    const int lane = tid & 31;
    const int gw   = blockIdx.x * 32 + wave;
    const int row0 = gw * 16;
    if (row0 >= nrows) return;               // wave-uniform: EXEC all-1 for WMMA

    // A-frag addressing (16x4 f32): lanes 0-15 -> K={k,k+1}, lanes 16-31 -> K={k+2,k+3}
    const int mrow  = row0 + (lane & 15);
    const int khalf = (lane >> 4) * 2;
    const int ncol  = lane & 15;
    const float* xrow = x + (size_t)mrow * F_IN;

    // per-lane B base: pair offset for this lane's K-half + column slot
    const float* bp = Wl + (khalf >> 1) * PAIR_STRIDE + ncol * 2;

    v8f c0 = {}, c1 = {}, c2 = {}, c3 = {};

    for (int k = 0; k < F_IN; k += 4) {
        v2f a = *(const v2f*)(xrow + k + khalf);

        v2f b0 = *(const v2f*)(bp +  0);     // cols  0..15  (tile 0)
        v2f b1 = *(const v2f*)(bp + 32);     // cols 16..31  (tile 1)
        v2f b2 = *(const v2f*)(bp + 64);     // cols 32..47  (tile 2)
        v2f b3 = *(const v2f*)(bp + 96);     // cols 48..63  (tile 3)
        bp += 2 * PAIR_STRIDE;               // advance 4 K values = 2 pairs

        c0 = __builtin_amdgcn_wmma_f32_16x16x4_f32(false, a, false, b0, (short)0, c0, false, false);
        c1 = __builtin_amdgcn_wmma_f32_16x16x4_f32(false, a, false, b1, (short)0, c1, false, false);
        c2 = __builtin_amdgcn_wmma_f32_16x16x4_f32(false, a, false, b2, (short)0, c2, false, false);
        c3 = __builtin_amdgcn_wmma_f32_16x16x4_f32(false, a, false, b3, (short)0, c3, false, false);
    }

    // C/D layout: VGPR r -> M = r (lanes0-15) / M = r+8 (lanes16-31), N = lane&15
    const int rbase = row0 + (lane >> 4) * 8;
    const int cbase = lane & 15;
    for (int r = 0; r < 8; ++r) {
        float* hp = h + (size_t)(rbase + r) * F_OUT + cbase;
        hp[ 0] = c0[r];
        hp[16] = c1[r];
        hp[32] = c2[r];
        hp[48] = c3[r];
    }
}

// ----------------------------------------------------------------- row init --
__global__ __launch_bounds__(256)
void gat_row_init(const float* __restrict__ h,
                  const float* __restrict__ a1, const float* __restrict__ a1b,
                  const float* __restrict__ a2, const float* __restrict__ a2b,
                  float* __restrict__ f1, float* __restrict__ f2,
                  float* __restrict__ mmax, float* __restrict__ ssum,
                  float* __restrict__ out, int n) {
    const int i = blockIdx.x * blockDim.x + threadIdx.x;
    if (i >= n) return;
    const float* hr = h + (size_t)i * F_OUT;
    float s1 = 0.f, s2 = 0.f;
    #pragma unroll 8
    for (int j = 0; j < F_OUT; ++j) {
        const float hv = hr[j];
        s1 += hv * a1[j];
        s2 += hv * a2[j];
    }
    f1[i] = s1 + a1b[0];
    f2[i] = s2 + a2b[0];
    mmax[i] = -INFINITY;
    ssum[i] = 0.f;
    float4 z = make_float4(0.f, 0.f, 0.f, 0.f);
    float4* o = (float4*)(out + (size_t)i * F_OUT);
    #pragma unroll
    for (int j = 0; j < F_OUT / 4; ++j) o[j] = z;
}

// portable float atomic-max (sign-split int trick; handles negatives & -inf)
__device__ __forceinline__ void atomic_max_f32(float* addr, float val) {
    if (val >= 0.f) atomicMax((int*)addr, __float_as_int(val));
    else            atomicMin((unsigned int*)addr, __float_as_uint(val));
}

// --------------------------------------------------------------- edge max --
__global__ __launch_bounds__(256)
void gat_edge_max(const int* __restrict__ src, const int* __restrict__ dst,
                  const float* __restrict__ adj,
                  const float* __restrict__ f1, const float* __restrict__ f2,
                  float* __restrict__ mmax, int ne) {
    const int e = blockIdx.x * blockDim.x + threadIdx.x;
    if (e >= ne) return;
    const int s_ = src[e];
    const int d_ = dst[e];
    const float av = adj[e];
    float v = av * f1[s_] + av * f2[d_];
    v = v > 0.f ? v : ALPHA * v;
    atomic_max_f32(&mmax[s_], v);
}

// ------------------------------------------------- edge exp + accumulate --
// one wave per edge: 32 lanes x 2 columns = 64-wide scatter into out[src].
__global__ __launch_bounds__(256)
void gat_edge_acc(const int* __restrict__ src, const int* __restrict__ dst,
                  const float* __restrict__ adj,
                  const float* __restrict__ f1, const float* __restrict__ f2,
                  const float* __restrict__ mmax, float* __restrict__ ssum,
                  const float* __restrict__ h, float* __restrict__ out, int ne) {
    const int lane = threadIdx.x & 31;
    const int e    = blockIdx.x * 8 + (threadIdx.x >> 5);
    if (e >= ne) return;                      // wave-uniform

    const int s_ = src[e];
    const int d_ = dst[e];
    const float av = adj[e];
    float v = av * f1[s_] + av * f2[d_];
    v = v > 0.f ? v : ALPHA * v;
    const float p = __expf(v - mmax[s_]);

    if (lane == 0) atomicAdd(&ssum[s_], p);   // softmax denominator

    const int col = lane * 2;
    const float* hp = h   + (size_t)d_ * F_OUT + col;
    float*       op = out + (size_t)s_ * F_OUT + col;
    __builtin_prefetch(hp, 0, 3);             // global_prefetch_b8 (h is L2-resident)
    atomicAdd(&op[0], p * hp[0]);             // L2 atomics: out (25.6MB) fits in 192MB L2
    atomicAdd(&op[1], p * hp[1]);
}

// ---------------------------------------------------------------- finalize --
__global__ __launch_bounds__(256)
void gat_finalize(float* __restrict__ out, const float* __restrict__ ssum,
                  const float* __restrict__ bias, int n) {
    const int i = blockIdx.x * blockDim.x + threadIdx.x;
    if (i >= n) return;
    const float sv  = ssum[i];
    const float inv = sv > 0.f ? 1.f / sv : 0.f;   // empty segment -> bias only
    float* o = out + (size_t)i * F_OUT;
    #pragma unroll 8
    for (int j = 0; j < F_OUT; ++j) o[j] = o[j] * inv + bias[j];
}

// ---------------------------------------------------------------------------
extern "C" void kernel_launch(void* const* d_in, const int* in_sizes, int n_in,
                              void* d_out, int out_size, void* d_ws, size_t ws_size,
                              hipStream_t stream) {
    const float* x    = (const float*)d_in[0];
    const int*   esrc = (const int*)  d_in[1];
    const int*   edst = (const int*)  d_in[2];
    const float* adj  = (const float*)d_in[3];
    const float* W    = (const float*)d_in[4];
    const float* a1   = (const float*)d_in[5];
    const float* a1b  = (const float*)d_in[6];
    const float* a2   = (const float*)d_in[7];
    const float* a2b  = (const float*)d_in[8];
    const float* bias = (const float*)d_in[9];
    float*       out  = (float*)d_out;

    const int n  = in_sizes[0] / F_IN;   // 100000
    const int ne = in_sizes[1];          // 3200000

    // workspace carve (floats): h[n*64] | f1[n] | f2[n] | m[n] | s[n]  (~27.2 MB)
    float* h  = (float*)d_ws;
    float* f1 = h  + (size_t)n * F_OUT;
    float* f2 = f1 + n;
    float* mm = f2 + n;
    float* ss = mm + n;

    const int tiles       = (n + 15) / 16;        // 6250 row tiles
    const int gemm_blocks = (tiles + 31) / 32;    // 32 waves/block

    gat_gemm_wmma<<<gemm_blocks, 1024, 0, stream>>>(x, W, h, n);
    gat_row_init<<<(n + 255) / 256, 256, 0, stream>>>(h, a1, a1b, a2, a2b,
                                                      f1, f2, mm, ss, out, n);
    gat_edge_max<<<(ne + 255) / 256, 256, 0, stream>>>(esrc, edst, adj, f1, f2, mm, ne);
    gat_edge_acc<<<(ne + 7) / 8, 256, 0, stream>>>(esrc, edst, adj, f1, f2,
                                                   mm, ss, h, out, ne);
    gat_finalize<<<(n + 255) / 256, 256, 0, stream>>>(out, ss, bias, n);
}